// FeatureSampler2D_60404420051267
// MI455X (gfx1250) — compile-verified
//
#include <hip/hip_runtime.h>
#include <hip/hip_bf16.h>

// Problem constants (fixed by the reference setup_inputs()).
#define PADDING 0.1f
#define EPSC    1e-3f
constexpr int B = 8, N = 65536, C = 64, H = 256, W = 256;
constexpr int HW = H * W;

typedef float v2f __attribute__((ext_vector_type(2)));
typedef float v4f __attribute__((ext_vector_type(4)));
typedef int   v4i __attribute__((vector_size(16)));     // matches builtin param type
typedef __attribute__((address_space(1))) v4i* gv4i_p;  // global (AS1), 128-bit
typedef __attribute__((address_space(3))) v4i* lv4i_p;  // LDS (AS3), 128-bit
typedef __attribute__((address_space(1))) int* gi32_p;  // global (AS1), 32-bit
typedef __attribute__((address_space(3))) int* li32_p;  // LDS (AS3), 32-bit

#if defined(__HIP_DEVICE_COMPILE__) && defined(__gfx1250__) && \
    __has_builtin(__builtin_amdgcn_global_load_async_to_lds_b128) && \
    __has_builtin(__builtin_amdgcn_s_wait_asynccnt)
#define USE_ASYNC_LD 1
#else
#define USE_ASYNC_LD 0
#endif

#if USE_ASYNC_LD && __has_builtin(__builtin_amdgcn_global_store_async_from_lds_b32)
#define USE_ASYNC_ST 1
#else
#define USE_ASYNC_ST 0
#endif

// ---------------------------------------------------------------------------
// Pass 1: transpose [B,C,H,W] -> channels-last ws layout [plane][B][H*W][C].
// Pure DMA-through-LDS: async b128 loads fill the tile (no VGPR data path),
// async b32 stores drain it in transposed order. Tile rows padded to 68
// floats so every 16B LDS chunk target stays 16B-aligned.
// ---------------------------------------------------------------------------
__global__ void fs2d_transpose_cl(const float* __restrict__ f_xz,
                                  const float* __restrict__ f_xy,
                                  const float* __restrict__ f_yz,
                                  float* __restrict__ ws) {
  __shared__ __align__(16) float tile[64][68];
  const int blk   = blockIdx.x;          // 3 * B * (HW/64) blocks
  const int pxblk = blk % (HW / 64);
  const int rem   = blk / (HW / 64);
  const int b     = rem % B;
  const int plane = rem / B;

  const float* src = (plane == 0) ? f_xz : (plane == 1) ? f_xy : f_yz;
  src += (size_t)b * C * HW;
  float* dst = ws + ((size_t)plane * B + b) * (size_t)HW * C;

  const int p0  = pxblk * 64;
  const int tid = threadIdx.x;           // 256 threads

  // ---- load phase: 1024 x b128 chunks (64 channels x 16 quads of pixels)
#if USE_ASYNC_LD
#pragma unroll
  for (int i = 0; i < 4; ++i) {
    const int chunk = i * 256 + tid;
    const int c = chunk >> 4;            // channel row 0..63
    const int q = chunk & 15;            // pixel quad 0..15
    const float* g = src + (size_t)c * HW + p0 + q * 4;
    __builtin_amdgcn_global_load_async_to_lds_b128(
        (gv4i_p)(void*)g, (lv4i_p)(void*)&tile[c][q * 4], 0, 0);
  }
  __builtin_amdgcn_s_wait_asynccnt(0);
  __syncthreads();
#else
#pragma unroll
  for (int i = 0; i < 4; ++i) {
    const int chunk = i * 256 + tid;
    const int c = chunk >> 4;
    const int q = chunk & 15;
    *(v4f*)&tile[c][q * 4] = *(const v4f*)(src + (size_t)c * HW + p0 + q * 4);
  }
  __syncthreads();
#endif

  // ---- store phase: 4096 x b32, transposed (coalesced along channels)
#if USE_ASYNC_ST
#pragma unroll
  for (int i = 0; i < 16; ++i) {
    const int e  = i * 256 + tid;
    const int px = e >> 6;               // pixel 0..63
    const int c  = e & 63;               // channel 0..63
    float* g = dst + (size_t)(p0 + px) * C + c;
    __builtin_amdgcn_global_store_async_from_lds_b32(
        (gi32_p)(void*)g, (li32_p)(void*)&tile[c][px], 0, 0);
  }
  __builtin_amdgcn_s_wait_asynccnt(0);
#else
#pragma unroll
  for (int i = 0; i < 16; ++i) {
    const int e  = i * 256 + tid;
    const int px = e >> 6;
    const int c  = e & 63;
    dst[(size_t)(p0 + px) * C + c] = tile[c][px];
  }
#endif
}

// ---------------------------------------------------------------------------
// Shared coordinate math (matches reference normalize + grid_sample,
// border padding, align_corners=True).
// ---------------------------------------------------------------------------
__device__ __forceinline__ void plane_coords(float a, float bb,
                                             int& x0, int& x1, int& y0, int& y1,
                                             float& w00, float& w01,
                                             float& w10, float& w11) {
  const float inv = 1.0f / (1.0f + PADDING + EPSC);
  float u = a  * inv + 0.5f;
  float v = bb * inv + 0.5f;
  u = fminf(fmaxf(u, 0.0f), 1.0f - EPSC);
  v = fminf(fmaxf(v, 0.0f), 1.0f - EPSC);
  // vg = 2u-1 ; x = (vg+1)*0.5*(W-1) = u*(W-1)
  const float x = u * (float)(W - 1);
  const float y = v * (float)(H - 1);
  const float xf = floorf(x), yf = floorf(y);
  const float wx = x - xf,    wy = y - yf;
  x0 = min(max((int)xf, 0), W - 1);
  x1 = min(x0 + 1, W - 1);
  y0 = min(max((int)yf, 0), H - 1);
  y1 = min(y0 + 1, H - 1);
  w00 = (1.0f - wy) * (1.0f - wx);
  w01 = (1.0f - wy) * wx;
  w10 = wy * (1.0f - wx);
  w11 = wy * wx;
}

// ---------------------------------------------------------------------------
// Pass 2 (channels-last path): one wave serves TWO points; lanes 0-15 hold
// point A (float4 channel quad per lane), lanes 16-31 point B. Every tap is
// one b128 wave load covering two contiguous 256B channel rows. Block's 16
// points (192B) are DMA'd into LDS via the CDNA5 async path (ASYNCcnt).
// ---------------------------------------------------------------------------
__global__ void fs2d_sample_cl(const float* __restrict__ points,
                               const float* __restrict__ ws,
                               float* __restrict__ out) {
  __shared__ __align__(16) float pts[64];   // 16 points x 3 floats used
  const int tid  = threadIdx.x;
  const int wave = tid >> 5;
  const int lane = tid & 31;
  const int gpi0 = blockIdx.x * 16;         // 16 points per block (8 waves x 2)

  // Prefetch next block's point slab into L2 (speculative, CDNA5 global_prefetch).
  __builtin_prefetch(points + (size_t)(gpi0 + 16) * 3, 0, 0);

#if USE_ASYNC_LD
  if (tid < 12) {  // 12 x b128 = 192 bytes, both sides 16B-aligned
    const float* g = points + (size_t)gpi0 * 3 + tid * 4;
    __builtin_amdgcn_global_load_async_to_lds_b128(
        (gv4i_p)(void*)g, (lv4i_p)(void*)&pts[tid * 4], 0, 0);
  }
  __builtin_amdgcn_s_wait_asynccnt(0);
  __syncthreads();
#else
  if (tid < 48) pts[tid] = points[(size_t)gpi0 * 3 + tid];
  __syncthreads();
#endif

  const int pt  = wave * 2 + (lane >> 4);   // 0..15 within block
  const int gpi = gpi0 + pt;
  const int b   = gpi >> 16;                // N == 65536
  const float p0 = pts[pt * 3 + 0];
  const float p1 = pts[pt * 3 + 1];
  const float p2 = pts[pt * 3 + 2];

  const int c4 = (lane & 15) * 4;           // channel quad within point
  const size_t planeStride = (size_t)B * HW * C;
  const float* pb = ws + (size_t)b * (size_t)HW * C;

  // plane order: xz(0,2), xy(0,1), yz(1,2) — matches reference accumulation order
  const float ua[3] = {p0, p0, p1};
  const float vb[3] = {p2, p1, p2};

  v4f acc = {0.0f, 0.0f, 0.0f, 0.0f};
#pragma unroll
  for (int pl = 0; pl < 3; ++pl) {
    int x0, x1, y0, y1; float w00, w01, w10, w11;
    plane_coords(ua[pl], vb[pl], x0, x1, y0, y1, w00, w01, w10, w11);
    const float* base = pb + (size_t)pl * planeStride;
    const v4f f00 = *(const v4f*)(base + (size_t)(y0 * W + x0) * C + c4);
    const v4f f01 = *(const v4f*)(base + (size_t)(y0 * W + x1) * C + c4);
    const v4f f10 = *(const v4f*)(base + (size_t)(y1 * W + x0) * C + c4);
    const v4f f11 = *(const v4f*)(base + (size_t)(y1 * W + x1) * C + c4);
    v4f s = f00 * w00;
    s += f01 * w01;
    s += f10 * w10;
    s += f11 * w11;
    acc += s;
  }

  // Output is streamed, never re-read: nontemporal 128-bit store.
  __builtin_nontemporal_store(acc, (v4f*)(out + (size_t)gpi * C + c4));
}

// ---------------------------------------------------------------------------
// Fallback (workspace too small): gather directly from [B,C,H,W] layout.
// ---------------------------------------------------------------------------
__global__ void fs2d_sample_direct(const float* __restrict__ points,
                                   const float* __restrict__ f_xz,
                                   const float* __restrict__ f_xy,
                                   const float* __restrict__ f_yz,
                                   float* __restrict__ out) {
  __shared__ __align__(16) float pts[32];
  const int tid  = threadIdx.x;
  const int wave = tid >> 5;
  const int lane = tid & 31;
  const int gpi0 = blockIdx.x * 8;
  __builtin_prefetch(points + (size_t)(gpi0 + 8) * 3, 0, 0);
  if (tid < 24) pts[tid] = points[(size_t)gpi0 * 3 + tid];
  __syncthreads();

  const int gpi = gpi0 + wave;
  const int b   = gpi >> 16;
  const float p0 = pts[wave * 3 + 0];
  const float p1 = pts[wave * 3 + 1];
  const float p2 = pts[wave * 3 + 2];

  const int c2 = lane * 2;
  const float* fp[3] = {f_xz, f_xy, f_yz};
  const float ua[3] = {p0, p0, p1};
  const float vb[3] = {p2, p1, p2};

  float accx = 0.0f, accy = 0.0f;
#pragma unroll
  for (int pl = 0; pl < 3; ++pl) {
    int x0, x1, y0, y1; float w00, w01, w10, w11;
    plane_coords(ua[pl], vb[pl], x0, x1, y0, y1, w00, w01, w10, w11);
    const float* base = fp[pl] + (size_t)b * C * HW;
    const float* ch0  = base + (size_t)c2 * HW;
    const float* ch1  = ch0 + HW;
    const int i00 = y0 * W + x0, i01 = y0 * W + x1;
    const int i10 = y1 * W + x0, i11 = y1 * W + x1;
    float sx = w00 * ch0[i00];
    sx = fmaf(w01, ch0[i01], sx);
    sx = fmaf(w10, ch0[i10], sx);
    sx = fmaf(w11, ch0[i11], sx);
    float sy = w00 * ch1[i00];
    sy = fmaf(w01, ch1[i01], sy);
    sy = fmaf(w10, ch1[i10], sy);
    sy = fmaf(w11, ch1[i11], sy);
    accx += sx; accy += sy;
  }
  v2f o; o.x = accx; o.y = accy;
  __builtin_nontemporal_store(o, (v2f*)(out + (size_t)gpi * C + c2));
}

extern "C" void kernel_launch(void* const* d_in, const int* in_sizes, int n_in,
                              void* d_out, int out_size, void* d_ws, size_t ws_size,
                              hipStream_t stream) {
  const float* points = (const float*)d_in[0];
  const float* f_xz   = (const float*)d_in[1];
  const float* f_xy   = (const float*)d_in[2];
  const float* f_yz   = (const float*)d_in[3];
  float* out = (float*)d_out;
  (void)in_sizes; (void)n_in; (void)out_size;

  const size_t needed = (size_t)3 * B * HW * C * sizeof(float);  // 384 MiB
  if (d_ws != nullptr && ws_size >= needed) {
    float* ws = (float*)d_ws;
    fs2d_transpose_cl<<<3 * B * (HW / 64), 256, 0, stream>>>(f_xz, f_xy, f_yz, ws);
    fs2d_sample_cl<<<(B * N) / 16, 256, 0, stream>>>(points, ws, out);
  } else {
    fs2d_sample_direct<<<(B * N) / 8, 256, 0, stream>>>(points, f_xz, f_xy, f_yz, out);
  }
}